// LearnBatchMMF_14577119002755
// MI455X (gfx1250) — compile-verified
//
#include <hip/hip_runtime.h>

#define NDIM 512
#define BDIM 32
#define LDIM 8
#define KDIM 8
#define KC   32          // k-chunk staged in LDS
#define NB   64          // n-tile per block
#define LDBP 160         // pair-row stride (floats): 160 mod 64 = 32 -> half-wave bank split

typedef float v2f __attribute__((ext_vector_type(2)));
typedef float v4f __attribute__((ext_vector_type(4)));
typedef float v8f __attribute__((ext_vector_type(8)));

// async-to-LDS builtin wants vector_size(16) int pointees, addrspace-qualified
typedef int v4i_vs __attribute__((vector_size(16)));
typedef __attribute__((address_space(1))) v4i_vs* gv4i_p;
typedef __attribute__((address_space(3))) v4i_vs* lv4i_p;

#if defined(__has_builtin)
#if __has_builtin(__builtin_amdgcn_global_load_async_to_lds_b128) && \
    __has_builtin(__builtin_amdgcn_s_wait_asynccnt)
#define HAVE_ASYNC_LDS 1
#endif
#endif

__device__ __forceinline__ void copy16_to_lds(const float* __restrict__ g, float* l) {
#ifdef HAVE_ASYNC_LDS
    __builtin_amdgcn_global_load_async_to_lds_b128(
        (gv4i_p)(void*)g,
        (lv4i_p)(void*)l, 0, 0);
#else
    *(v4f*)l = *(const v4f*)g;
#endif
}

__device__ __forceinline__ void drain_async_lds() {
#ifdef HAVE_ASYNC_LDS
    __builtin_amdgcn_s_wait_asynccnt(0);
#endif
}

// ---------------------------------------------------------------------------
// 1) Copy A -> workspace, init right = I (right lives in its output slot)
// ---------------------------------------------------------------------------
__global__ void init_kernel(const float* __restrict__ Ain,
                            float* __restrict__ Awork,
                            float* __restrict__ right) {
    size_t i = (size_t)blockIdx.x * blockDim.x + threadIdx.x;
    const size_t total = (size_t)BDIM * NDIM * NDIM;
    if (i >= total) return;
    Awork[i] = Ain[i];
    int rc = (int)(i % ((size_t)NDIM * NDIM));
    int r = rc >> 9, c = rc & (NDIM - 1);
    right[i] = (r == c) ? 1.0f : 0.0f;
}

// ---------------------------------------------------------------------------
// 2) Sparse MMF layer loop: one block per batch, 8 layers sequentially.
//    Row gathers (contiguous 512-float rows) use async global->LDS DMA.
// ---------------------------------------------------------------------------
__global__ void __launch_bounds__(256)
layers_kernel(const float* __restrict__ O_stack,
              const int*   __restrict__ indices,
              float* __restrict__ Awork,
              float* __restrict__ right) {
    const int b   = blockIdx.x;
    const int tid = threadIdx.x;
    __shared__ float lds[KDIM][NDIM];      // 16 KB staging buffer
    __shared__ float sO[KDIM * KDIM];
    __shared__ int   sidx[KDIM];

    float* A = Awork + (size_t)b * NDIM * NDIM;
    float* R = right + (size_t)b * NDIM * NDIM;

    for (int l = 0; l < LDIM; ++l) {
        if (tid < KDIM)        sidx[tid] = indices[(b * LDIM + l) * KDIM + tid];
        if (tid < KDIM * KDIM) sO[tid]   = O_stack[((size_t)l * BDIM + b) * KDIM * KDIM + tid];
        __syncthreads();

        // ---- rows of A (async DMA: 8 contiguous rows -> LDS)
        for (int e = tid; e < KDIM * NDIM / 4; e += blockDim.x) {
            int i = e >> 7, c4 = (e & 127) * 4;
            copy16_to_lds(A + (size_t)sidx[i] * NDIM + c4, &lds[i][c4]);
        }
        drain_async_lds();
        __syncthreads();
        for (int e = tid; e < KDIM * NDIM; e += blockDim.x) {
            int i = e >> 9, c = e & 511;
            float s = 0.f;
#pragma unroll
            for (int j = 0; j < KDIM; ++j) s += sO[i * KDIM + j] * lds[j][c];
            A[(size_t)sidx[i] * NDIM + c] = s;
        }
        __threadfence();
        __syncthreads();

        // ---- cols of A (strided gather, scalar loads)
        for (int e = tid; e < KDIM * NDIM; e += blockDim.x)
            lds[e >> 9][e & 511] = A[(size_t)(e & 511) * NDIM + sidx[e >> 9]];
        __syncthreads();
        for (int e = tid; e < KDIM * NDIM; e += blockDim.x) {
            int i = e >> 9, r = e & 511;
            float s = 0.f;
#pragma unroll
            for (int j = 0; j < KDIM; ++j) s += sO[i * KDIM + j] * lds[j][r];
            A[(size_t)r * NDIM + sidx[i]] = s;
        }
        __threadfence();
        __syncthreads();

        // ---- rows of right (async DMA)
        for (int e = tid; e < KDIM * NDIM / 4; e += blockDim.x) {
            int i = e >> 7, c4 = (e & 127) * 4;
            copy16_to_lds(R + (size_t)sidx[i] * NDIM + c4, &lds[i][c4]);
        }
        drain_async_lds();
        __syncthreads();
        for (int e = tid; e < KDIM * NDIM; e += blockDim.x) {
            int i = e >> 9, c = e & 511;
            float s = 0.f;
#pragma unroll
            for (int j = 0; j < KDIM; ++j) s += sO[i * KDIM + j] * lds[j][c];
            R[(size_t)sidx[i] * NDIM + c] = s;
        }
        __threadfence();
        __syncthreads();
    }
}

// ---------------------------------------------------------------------------
// 3) D = A .* (active outer active), diagonal forced to A's diagonal
// ---------------------------------------------------------------------------
__global__ void __launch_bounds__(256)
mask_kernel(const float* __restrict__ Awork,
            const int*   __restrict__ wavelet,
            float* __restrict__ Dout) {
    const int b = blockIdx.x;
    const int rowBase = blockIdx.y * 32;
    __shared__ unsigned char act[NDIM];
    for (int e = threadIdx.x; e < NDIM; e += blockDim.x) act[e] = 1;
    __syncthreads();
    if (threadIdx.x < LDIM) act[wavelet[b * LDIM + threadIdx.x]] = 0;
    __syncthreads();

    const float* A = Awork + (size_t)b * NDIM * NDIM;
    float* Dd = Dout + (size_t)b * NDIM * NDIM;
    for (int e = threadIdx.x; e < 32 * NDIM; e += blockDim.x) {
        int r = rowBase + (e >> 9);
        int c = e & 511;
        float v = A[(size_t)r * NDIM + c];
        Dd[(size_t)r * NDIM + c] = (r == c) ? v : ((act[r] & act[c]) ? v : 0.0f);
    }
}

// ---------------------------------------------------------------------------
// 4/5) Batched fp32 GEMM via V_WMMA_F32_16X16X4_F32, LDS-staged B panel in a
//      PAIR-INTERLEAVED layout: element (k,n) at (k>>1)*LDBP + 2n + (k&1).
//      A B-fragment {B[k][n], B[k+1][n]} is then one contiguous ds_load_b64
//      straight into an even VGPR pair (no repack movs); the two half-waves
//      (pair rows p and p+1, 160*4B apart -> +32 banks) are conflict-free.
//      Block = 8 waves -> 128x64 C tile; B panel shared by all 8 waves.
//      Grid: (M/128, N/64, B).
// ---------------------------------------------------------------------------
template <bool TRANSA>
__global__ void __launch_bounds__(256)
gemm_kernel(const float* __restrict__ Ag,
            const float* __restrict__ Bg,
            float* __restrict__ Cg) {
    const int b      = blockIdx.z;
    const int wave   = threadIdx.x >> 5;
    const int lane   = threadIdx.x & 31;
    const int laneLo = lane & 15;
    const int laneHi = lane >> 4;          // 0: K pair {k,k+1}, 1: K pair {k+2,k+3}
    const int m0     = blockIdx.x * 128 + wave * 16;
    const int n0     = blockIdx.y * NB;

    const float* A  = Ag + (size_t)b * NDIM * NDIM;
    const float* Bm = Bg + (size_t)b * NDIM * NDIM;
    float*       C  = Cg + (size_t)b * NDIM * NDIM;

    __shared__ float ldsB[(KC / 2) * LDBP];   // 16 pair-rows x 160 = 10 KB

    // staging map: thread t fills rows (t>>4) and (t>>4)+16, 4 floats each
    const int srow = threadIdx.x >> 4;
    const int scol = (threadIdx.x & 15) * 4;

    v8f acc0 = {}, acc1 = {}, acc2 = {}, acc3 = {};

    for (int kc = 0; kc < NDIM; kc += KC) {
        // ---- stage B[kc..kc+KC) x [n0..n0+NB) into interleaved LDS panel
#pragma unroll
        for (int h = 0; h < 2; ++h) {
            const int r = srow + 16 * h;
            v4f t = *(const v4f*)(Bm + (size_t)(kc + r) * NDIM + n0 + scol);
            float* dst = ldsB + (r >> 1) * LDBP + 2 * scol + (r & 1);
            dst[0] = t.x; dst[2] = t.y; dst[4] = t.z; dst[6] = t.w;
        }
        __syncthreads();

#pragma unroll
        for (int k0 = 0; k0 < KC; k0 += 4) {
            const int kg = kc + k0;
            // A-fragment 16x4: lanes 0-15 hold K={k,k+1}, lanes 16-31 K={k+2,k+3}
            v2f a;
            if (!TRANSA) {
                a = *(const v2f*)(A + (size_t)(m0 + laneLo) * NDIM + (kg + 2 * laneHi));
            } else {
                a.x = A[(size_t)(kg + 2 * laneHi)     * NDIM + (m0 + laneLo)];
                a.y = A[(size_t)(kg + 2 * laneHi + 1) * NDIM + (m0 + laneLo)];
            }
            // B-fragments: one ds_load_b64 each, already in WMMA pairing
            const float* bp = ldsB + ((k0 >> 1) + laneHi) * LDBP + 2 * laneLo;
            v2f b0 = *(const v2f*)(bp + 0);
            v2f b1 = *(const v2f*)(bp + 32);
            v2f b2 = *(const v2f*)(bp + 64);
            v2f b3 = *(const v2f*)(bp + 96);

            acc0 = __builtin_amdgcn_wmma_f32_16x16x4_f32(false, a, false, b0, (short)0, acc0, false, false);
            acc1 = __builtin_amdgcn_wmma_f32_16x16x4_f32(false, a, false, b1, (short)0, acc1, false, false);
            acc2 = __builtin_amdgcn_wmma_f32_16x16x4_f32(false, a, false, b2, (short)0, acc2, false, false);
            acc3 = __builtin_amdgcn_wmma_f32_16x16x4_f32(false, a, false, b3, (short)0, acc3, false, false);
        }
        __syncthreads();
    }

    // C/D layout: VGPR r -> row m0 + r + 8*laneHi, col = lane within 16
#pragma unroll
    for (int r = 0; r < 8; ++r) {
        const size_t row = (size_t)(m0 + r + 8 * laneHi) * NDIM + n0 + laneLo;
        C[row]      = acc0[r];
        C[row + 16] = acc1[r];
        C[row + 32] = acc2[r];
        C[row + 48] = acc3[r];
    }
}

// ---------------------------------------------------------------------------
// Launch: init -> layers -> mask -> T = D@right -> A_rec = right^T @ T
// d_out = [A_rec | right | D], d_ws = [A_work | T]  (64 MB)
// ---------------------------------------------------------------------------
extern "C" void kernel_launch(void* const* d_in, const int* in_sizes, int n_in,
                              void* d_out, int out_size, void* d_ws, size_t ws_size,
                              hipStream_t stream) {
    const float* A_in    = (const float*)d_in[0];
    const float* O_stack = (const float*)d_in[1];
    const int*   indices = (const int*)d_in[2];
    const int*   wavelet = (const int*)d_in[3];

    float* out = (float*)d_out;
    const size_t mat = (size_t)BDIM * NDIM * NDIM;
    float* A_rec = out;
    float* right = out + mat;
    float* Dout  = out + 2 * mat;

    float* Awork = (float*)d_ws;
    float* T     = Awork + mat;

    const int threads = 256;
    init_kernel<<<dim3((unsigned)((mat + threads - 1) / threads)), threads, 0, stream>>>(A_in, Awork, right);
    layers_kernel<<<dim3(BDIM), threads, 0, stream>>>(O_stack, indices, Awork, right);
    mask_kernel<<<dim3(BDIM, NDIM / 32), threads, 0, stream>>>(Awork, wavelet, Dout);

    dim3 ggrid(NDIM / 128, NDIM / NB, BDIM);   // (4, 8, 32)
    gemm_kernel<false><<<ggrid, threads, 0, stream>>>(Dout, right, T);
    gemm_kernel<true><<<ggrid, threads, 0, stream>>>(right, T, A_rec);
}